// LiquidEchoModel_75892072121051
// MI455X (gfx1250) — compile-verified
//
#include <hip/hip_runtime.h>
#include <hip/hip_bf16.h>
#include <math.h>
#include <stdint.h>

// ---------------------------------------------------------------------------
// LiquidEchoModel on MI455X (gfx1250, wave32).
//  Phase 1: sequential recurrence (64 steps x 2 layers) as a chain of small
//           kernels; state in d_ws; LDS-resident 4096-entry sin/cos LUT
//           reproduces the reference's quantized lut_sin_cos. The
//           step-invariant 1/(1+|res_w|) is precomputed once (divide -> FMA).
//  Phase 2: deferred logits projection as ONE fp32 GEMM
//           (256 x 512) @ (512 x 50257) using V_WMMA_F32_16X16X4_F32.
//           proj_r/proj_i (103 MB) are L2-resident -> bandwidth-bound, so
//           full-precision f32 WMMA is the right precision choice. The A slab
//           is staged to LDS by the Tensor Data Mover (padded tile ->
//           bank-conflict-free 516-float stride), fallback to manual staging.
// ---------------------------------------------------------------------------

typedef float f2  __attribute__((ext_vector_type(2)));
typedef float f4  __attribute__((ext_vector_type(4)));
typedef float v8f __attribute__((ext_vector_type(8)));
typedef unsigned int v4u __attribute__((ext_vector_type(4)));
typedef int v4i __attribute__((ext_vector_type(4)));
typedef int v8i __attribute__((ext_vector_type(8)));

#define DEV __device__ __forceinline__

constexpr int B = 4, S = 64, V = 50257, D = 256, N = 512, L = 2, H = 2;
constexpr int LUT = 4096;
constexpr float PHI_F      = 1.6180339887498948f;
constexpr float LUT_SCALE  = 651.89864690440330f;     // 4096 / (2*pi)
constexpr float GRID_STEP  = 1.5339807878856412e-3f;  // 2*pi / 4096
constexpr float INV_SCALE  = 1.0f / 16.0f;            // 1/sqrt(D)

// workspace layout (float offsets); total ~1.65 MB
constexpr int WS_HR   = 0;                  // (B,D)
constexpr int WS_HI   = WS_HR + B * D;      // (B,D)
constexpr int WS_XC   = WS_HI + B * D;      // (B,D)
constexpr int WS_CS   = WS_XC + B * D;      // (B,N)
constexpr int WS_SS   = WS_CS + B * N;      // (B,N)
constexpr int WS_MR   = WS_SS + B * N;      // (L,H,B,D)
constexpr int WS_MI   = WS_MR + L * H * B * D;
constexpr int WS_RR   = WS_MI + L * H * B * D;  // (B,D)
constexpr int WS_RI   = WS_RR + B * D;          // (B,D)
constexpr int WS_HCAT = WS_RI + B * D;          // (S*B, 2D)
constexpr int WS_IRW  = WS_HCAT + S * B * 2 * D; // (L,N,D) = 1/(1+|res_w|)

DEV int lut_idx(float th) {
    // round-half-even then floor-mod 4096 (power of 2 -> AND works for negatives)
    return ((int)rintf(th * LUT_SCALE)) & (LUT - 1);
}

DEV float wave_reduce(float v) {
    #pragma unroll
    for (int off = 16; off > 0; off >>= 1) v += __shfl_xor(v, off, 32);
    return v;
}

DEV float silu(float x) { return x / (1.0f + expf(-x)); }

// ---- once-per-launch precompute -------------------------------------------

__global__ __launch_bounds__(256)
void k_prep(const float* __restrict__ rw, float* __restrict__ ws) {
    int i = blockIdx.x * 256 + threadIdx.x;        // L*N*D = 262144 total
    ws[WS_IRW + i] = 1.0f / (1.0f + fabsf(rw[i]));
}

// ---- step kernels ---------------------------------------------------------

__global__ __launch_bounds__(256)
void k_embed(const int* __restrict__ ids, const float* __restrict__ emb,
             float* __restrict__ ws, int s) {
    for (int i = threadIdx.x; i < B * D; i += 256) {
        int b = i >> 8, d = i & (D - 1);
        long row = (long)ids[b * S + s];
        ws[WS_HR + i] = emb[row * (2 * D) + d];
        ws[WS_HI + i] = emb[row * (2 * D) + D + d];
    }
}

// xc[b,dp] = concat(h_r,h_i)[b] . collapse_w[l][dp] + collapse_b[l][dp]
__global__ __launch_bounds__(256)
void k_xc(const float* __restrict__ cw, const float* __restrict__ cb,
          float* __restrict__ ws, int l) {
    __shared__ float cat[2 * D];
    const int b = blockIdx.x, dp = threadIdx.x;
    for (int i = threadIdx.x; i < 2 * D; i += 256)
        cat[i] = (i < D) ? ws[WS_HR + b * D + i] : ws[WS_HI + b * D + (i - D)];
    __syncthreads();
    const float* wrow = cw + ((size_t)(l * D + dp)) * (2 * D);
    float acc = cb[l * D + dp];
    #pragma unroll 4
    for (int j = 0; j < 2 * D; j += 4) {
        f4 wv = *(const f4*)(wrow + j);
        acc += wv.x * cat[j] + wv.y * cat[j + 1] + wv.z * cat[j + 2] + wv.w * cat[j + 3];
    }
    ws[WS_XC + b * D + dp] = acc;
}

// one wave per n: cos/sin_sum[b,n] for all 4 b's, sharing the n-row loads
__global__ __launch_bounds__(256)
void k_theta(const float* __restrict__ rb,
             const float* __restrict__ ac, const float* __restrict__ asn,
             float* __restrict__ ws, int l, float t) {
    __shared__ float lutS[LUT], lutC[LUT];
    __shared__ float xcL[B * D];
    for (int i = threadIdx.x; i < LUT; i += 256) {
        float sv, cv; sincosf((float)i * GRID_STEP, &sv, &cv);
        lutS[i] = sv; lutC[i] = cv;
    }
    for (int i = threadIdx.x; i < B * D; i += 256) xcL[i] = ws[WS_XC + i];
    __syncthreads();
    const int lane = threadIdx.x & 31;
    const int n = blockIdx.x * 8 + (threadIdx.x >> 5);   // 512 waves == N
    const size_t base = ((size_t)(l * N + n)) * D;
    const float* irw = ws + WS_IRW;
    float cs0 = 0, cs1 = 0, cs2 = 0, cs3 = 0;
    float ss0 = 0, ss1 = 0, ss2 = 0, ss3 = 0;
    for (int d = lane; d < D; d += 32) {
        float iw  = irw[base + d];              // precomputed 1/(1+|res_w|)
        float rbv = rb[base + d] + t;
        float acv = ac[base + d];
        float asv = asn[base + d];
        int i0 = lut_idx(fmaf(xcL[0 * D + d], iw, rbv));
        int i1 = lut_idx(fmaf(xcL[1 * D + d], iw, rbv));
        int i2 = lut_idx(fmaf(xcL[2 * D + d], iw, rbv));
        int i3 = lut_idx(fmaf(xcL[3 * D + d], iw, rbv));
        cs0 += lutC[i0] * acv; ss0 += lutS[i0] * asv;
        cs1 += lutC[i1] * acv; ss1 += lutS[i1] * asv;
        cs2 += lutC[i2] * acv; ss2 += lutS[i2] * asv;
        cs3 += lutC[i3] * acv; ss3 += lutS[i3] * asv;
    }
    cs0 = wave_reduce(cs0); ss0 = wave_reduce(ss0);
    cs1 = wave_reduce(cs1); ss1 = wave_reduce(ss1);
    cs2 = wave_reduce(cs2); ss2 = wave_reduce(ss2);
    cs3 = wave_reduce(cs3); ss3 = wave_reduce(ss3);
    if (lane == 0) {
        ws[WS_CS + 0 * N + n] = cs0; ws[WS_SS + 0 * N + n] = ss0;
        ws[WS_CS + 1 * N + n] = cs1; ws[WS_SS + 1 * N + n] = ss1;
        ws[WS_CS + 2 * N + n] = cs2; ws[WS_SS + 2 * N + n] = ss2;
        ws[WS_CS + 3 * N + n] = cs3; ws[WS_SS + 3 * N + n] = ss3;
    }
}

// r_r/r_i[b,d] = silu(cos_sum[b] . out_r_w[l][d]) etc.
__global__ __launch_bounds__(256)
void k_out(const float* __restrict__ orw, const float* __restrict__ oiw,
           float* __restrict__ ws, int l) {
    __shared__ float csL[N], ssL[N];
    const int b = blockIdx.x, d = threadIdx.x;
    for (int i = threadIdx.x; i < N; i += 256) {
        csL[i] = ws[WS_CS + b * N + i];
        ssL[i] = ws[WS_SS + b * N + i];
    }
    __syncthreads();
    const float* rrow = orw + ((size_t)(l * D + d)) * N;
    const float* irow = oiw + ((size_t)(l * D + d)) * N;
    float ar = 0.f, ai = 0.f;
    #pragma unroll 4
    for (int nn = 0; nn < N; nn += 4) {
        f4 wr = *(const f4*)(rrow + nn); f4 wi = *(const f4*)(irow + nn);
        ar += wr.x * csL[nn] + wr.y * csL[nn + 1] + wr.z * csL[nn + 2] + wr.w * csL[nn + 3];
        ai += wi.x * ssL[nn] + wi.y * ssL[nn + 1] + wi.z * ssL[nn + 2] + wi.w * ssL[nn + 3];
    }
    ws[WS_RR + b * D + d] = silu(ar);
    ws[WS_RI + b * D + d] = silu(ai);
}

// head memory update + gating; writes new h, and Hcat row on the last layer
__global__ __launch_bounds__(1024)
void k_heads(const float* __restrict__ wq, const float* __restrict__ bq,
             const float* __restrict__ wo, const float* __restrict__ bo,
             const float* __restrict__ gate, float* __restrict__ ws,
             int l, float t, int s) {
    __shared__ float lutS[LUT], lutC[LUT];
    __shared__ float wIr[32], wIi[32];
    for (int i = threadIdx.x; i < LUT; i += 1024) {
        float sv, cv; sincosf((float)i * GRID_STEP, &sv, &cv);
        lutS[i] = sv; lutC[i] = cv;
    }
    const int tid = threadIdx.x, b = tid >> 8, d = tid & (D - 1);
    const int lane = tid & 31, wvid = tid >> 5;
    const float hr = ws[WS_HR + b * D + d];
    const float hi = ws[WS_HI + b * D + d];
    const float tphi = t * PHI_F;
    float gr = 0.f, gi = 0.f;
    __syncthreads();
    for (int hh = 0; hh < H; ++hh) {
        const size_t mo = ((size_t)((l * H + hh) * B + b)) * D + d;
        const float mr = ws[WS_MR + mo], mi = ws[WS_MI + mo];
        const size_t qo = (size_t)(l * H + hh) * D + d;
        float arg = hr / (1.0f + fabsf(wq[qo])) + bq[qo];
        int idx = lut_idx(arg);
        float sq = lutS[idx], cq = lutC[idx];
        float pir = wave_reduce(cq * mr + sq * mi);
        float pii = wave_reduce(cq * mi - sq * mr);
        __syncthreads();
        if (lane == 0) { wIr[wvid] = pir; wIi[wvid] = pii; }
        __syncthreads();
        float ir = 0.f, ii = 0.f;     // deterministic fixed-order tree
        #pragma unroll
        for (int j = 0; j < 8; ++j) { ir += wIr[(b << 3) + j]; ii += wIi[(b << 3) + j]; }
        float mag = sqrtf(ir * ir + ii * ii);
        float sg = 1.0f / (1.0f + expf(-(mag * INV_SCALE - 2.0f)));
        float alpha = expf(-(1.0f - sg));                 // K = 1.0
        float br = alpha * hr + (1.0f - alpha) * mr;
        float bi = alpha * hi + (1.0f - alpha) * mi;
        float wlo = 1.0f + fabsf(wo[qo]);
        int i1 = lut_idx(br / wlo + bo[qo] + tphi);
        int i2 = lut_idx(bi / wlo + bo[qo] + tphi);
        float sor = lutS[i1], cor = lutC[i1];
        float soi = lutS[i2], coi = lutC[i2];
        float er = cor * coi - sor * soi;
        float ei = cor * soi + sor * coi;
        ws[WS_MR + mo] = er; ws[WS_MI + mo] = ei;
        gr += er; gi += ei;
    }
    gr *= (1.0f / H); gi *= (1.0f / H);
    const float g = gate[l];
    float nhr = ws[WS_RR + b * D + d] * (1.0f + g * gr);
    float nhi = ws[WS_RI + b * D + d] * (1.0f + g * gi);
    ws[WS_HR + b * D + d] = nhr;
    ws[WS_HI + b * D + d] = nhi;
    if (l == L - 1) {
        size_t mrow = (size_t)(s * B + b) * (2 * D);
        ws[WS_HCAT + mrow + d]     = nhr;
        ws[WS_HCAT + mrow + D + d] = nhi;
    }
}

// ---- phase 2: logits = Hcat (256x512) @ [proj_r|proj_i]^T via f32 WMMA ----
// LDS layout: row stride 516 floats, with +2 floats padding after the first
// 256 columns of each row (matches TDM pad_interval=256dw / pad_amount=2dw).
// Bank math: (516 % 64 == 4) -> all 32 lanes' b64 A-reads hit distinct banks.
__global__ __launch_bounds__(256)
void k_gemm(const float* __restrict__ hc, const float* __restrict__ pr,
            const float* __restrict__ pi, float* __restrict__ out) {
    constexpr int AST = 516;
    __shared__ float Alds[16 * AST];
    const int tid = threadIdx.x;
    const int m0  = blockIdx.y * 16;
#if __has_builtin(__builtin_amdgcn_tensor_load_to_lds) && __has_builtin(__builtin_amdgcn_s_wait_tensorcnt)
    if (tid < 32) {  // wave 0 drives the Tensor Data Mover for the A slab
        unsigned lds_base = (unsigned)(uintptr_t)(void*)&Alds[0];
        unsigned long long ga =
            (unsigned long long)(uintptr_t)(hc + (size_t)m0 * 512);
        v4u g0; v8i g1;
        v4i g2 = {0, 0, 0, 0}, g3 = {0, 0, 0, 0};
        v8i g4 = {0, 0, 0, 0, 0, 0, 0, 0};            // 6-arg toolchain variant
        g0[0] = 1u;                                   // count=1, user descriptor
        g0[1] = lds_base;                             // lds_addr (bytes)
        g0[2] = (unsigned)(ga & 0xffffffffu);         // global_addr[31:0]
        g0[3] = (unsigned)((ga >> 32) & 0x01ffffffu)  // global_addr[56:32]
                | (2u << 30);                         // type=2 ("image")
        g1[0] = (int)((2u << 16)    // data_size=4B
                | (1u << 20)        // pad_enable
                | (7u << 22)        // pad_interval: 256 DWORDs
                | (1u << 25));      // pad_amount: 2 DWORDs
        g1[1] = (int)((512u & 0xffffu) << 16);        // tensor_dim0[15:0]=512
        g1[2] = (int)(((512u >> 16) & 0xffffu) |      // tensor_dim0[31:16]
                      ((16u & 0xffffu) << 16));       // tensor_dim1[15:0]=16
        g1[3] = (int)(((16u >> 16) & 0xffffu) |       // tensor_dim1[31:16]
                      (512u << 16));                  // tile_dim0=512
        g1[4] = (int)(16u);                           // tile_dim1=16, tile_dim2=0
        g1[5] = (int)512;                             // tensor_dim0_stride[31:0]
        g1[6] = 0;                                    // stride0 hi / stride1 lo
        g1[7] = 0;
        __builtin_amdgcn_tensor_load_to_lds(g0, g1, g2, g3, g4, 0);
        __builtin_amdgcn_s_wait_tensorcnt(0);
    }
#else
    for (int i = tid; i < 16 * 256; i += 256) {       // manual padded staging
        int row = i >> 8, c2 = (i & 255) << 1;
        f2 v2 = *(const f2*)(hc + (size_t)(m0 + row) * 512 + c2);
        *(f2*)(&Alds[row * AST + c2 + (c2 >= 256 ? 2 : 0)]) = v2;
    }
#endif
    __syncthreads();
    const int lane = tid & 31, wv = tid >> 5;
    const int nt = blockIdx.x * 8 + wv;      // 16-wide N tile per wave
    const int n  = lane & 15, kh = lane >> 4;
    const int v  = nt * 16 + n;
    const int vc = (v < V) ? v : 0;
    const float* rowr = pr + (size_t)vc * 256;
    const float* rowi = pi + (size_t)vc * 256;
    __builtin_prefetch(rowr, 0, 0);          // -> global_prefetch_b8
    __builtin_prefetch(rowi, 0, 0);
    v8f acc = {0.f, 0.f, 0.f, 0.f, 0.f, 0.f, 0.f, 0.f};
    // half 1: K in [0,256) -> proj_r ; clean affine address streams
    #pragma unroll 4
    for (int k = 0; k < 256; k += 4) {
        int kk = k + 2 * kh;
        f2 a   = *(const f2*)(&Alds[n * AST + kk]);
        f2 bfr = *(const f2*)(rowr + kk);
        acc = __builtin_amdgcn_wmma_f32_16x16x4_f32(
                  false, a, false, bfr, (short)0, acc, false, false);
    }
    // half 2: K in [256,512) -> proj_i ; LDS offset +258 = 256 data + 2 pad
    #pragma unroll 4
    for (int k = 0; k < 256; k += 4) {
        int kk = k + 2 * kh;
        f2 a   = *(const f2*)(&Alds[n * AST + 258 + kk]);
        f2 bfr = *(const f2*)(rowi + kk);
        acc = __builtin_amdgcn_wmma_f32_16x16x4_f32(
                  false, a, false, bfr, (short)0, acc, false, false);
    }
    if (v < V) {
        #pragma unroll
        for (int j = 0; j < 8; ++j) {        // D: VGPR j -> M = j + 8*half
            int m  = m0 + j + 8 * kh;        // m = s*B + b
            int ss = m >> 2, bb = m & 3;
            out[((size_t)(bb * S + ss)) * V + v] = acc[j];
        }
    }
}

// ---------------------------------------------------------------------------

extern "C" void kernel_launch(void* const* d_in, const int* in_sizes, int n_in,
                              void* d_out, int out_size, void* d_ws, size_t ws_size,
                              hipStream_t stream) {
    (void)in_sizes; (void)n_in; (void)out_size; (void)ws_size;
    const int*   ids = (const int*)  d_in[0];
    const float* emb = (const float*)d_in[1];
    const float* cw  = (const float*)d_in[2];
    const float* cb  = (const float*)d_in[3];
    const float* rw  = (const float*)d_in[4];
    const float* rb  = (const float*)d_in[5];
    const float* ac  = (const float*)d_in[6];
    const float* asn = (const float*)d_in[7];
    const float* orw = (const float*)d_in[8];
    const float* oiw = (const float*)d_in[9];
    const float* wq  = (const float*)d_in[10];
    const float* bq  = (const float*)d_in[11];
    const float* wo  = (const float*)d_in[12];
    const float* bo  = (const float*)d_in[13];
    const float* gate= (const float*)d_in[14];
    const float* pr  = (const float*)d_in[15];
    const float* pi  = (const float*)d_in[16];
    float* ws  = (float*)d_ws;
    float* out = (float*)d_out;

    // zero the recurrent memory (graph-capturable memset node)
    (void)hipMemsetAsync((char*)d_ws + (size_t)WS_MR * sizeof(float), 0,
                         (size_t)(2 * L * H * B * D) * sizeof(float), stream);
    // precompute 1/(1+|res_w|) once (removes 64M fp32 divides from the scan)
    k_prep<<<(L * N * D) / 256, 256, 0, stream>>>(rw, ws);

    for (int s = 0; s < S; ++s) {
        float t = (float)s;
        k_embed<<<1, 256, 0, stream>>>(ids, emb, ws, s);
        for (int l = 0; l < L; ++l) {
            k_xc   <<<B, 256, 0, stream>>>(cw, cb, ws, l);
            k_theta<<<N / 8, 256, 0, stream>>>(rb, ac, asn, ws, l, t);
            k_out  <<<B, 256, 0, stream>>>(orw, oiw, ws, l);
            k_heads<<<1, 1024, 0, stream>>>(wq, bq, wo, bo, gate, ws, l, t, s);
        }
    }
    dim3 grid((V + 16 * 8 - 1) / (16 * 8), (S * B) / 16);   // (393, 16)
    k_gemm<<<grid, 256, 0, stream>>>(ws + WS_HCAT, pr, pi, out);
}